// TransitionDown_68178310857445
// MI455X (gfx1250) — compile-verified
//
#include <hip/hip_runtime.h>
#include <stdint.h>

// ---------------- problem constants ----------------
#define BATCH   4
#define N_PER   8192
#define M_PER   2048            // N_PER / 4
#define KNN     16
#define C_IN    64
#define C_OUT   128
#define C_G     67              // 3 + C_IN
#define C_PAD   68              // padded to multiple of 4 for 16x16x4 WMMA
#define NCHUNK  17              // C_PAD / 4
#define EPSBN   1e-5f
#define M_TOT   (BATCH * M_PER) // 8192 sampled points
#define MPB     8               // m-tiles per GEMM block
#define GEMM_GRID (M_TOT / MPB) // 1024

typedef float v2f __attribute__((ext_vector_type(2)));
typedef float v8f __attribute__((ext_vector_type(8)));
typedef unsigned int v4u __attribute__((ext_vector_type(4)));
typedef int v4i __attribute__((ext_vector_type(4)));
typedef int v8i __attribute__((ext_vector_type(8)));

// ---------------- workspace layout (bytes) ----------------
#define WS_NIDX_OFF   0           // int32 [M_TOT*KNN] = 512 KB
#define WS_STATS_OFF  524288      // float[256] (sum[128], sumsq[128])
#define WS_SCALE_OFF  525312      // float[128]
#define WS_BIAS_OFF   525824      // float[128]

// ---------------- output layout (float elements in d_out) ----------------
#define OUT_NEWP_OFF  0
#define OUT_FEAT_OFF  (M_TOT * 3)
#define OUT_RS_OFF    (M_TOT * 3 + M_TOT * C_OUT)

// Tensor Data Mover availability (this toolchain: clang-23 6-arg form).
// __has_builtin is false in the host pass, so the fallback compiles there.
#if __has_builtin(__builtin_amdgcn_tensor_load_to_lds) && \
    __has_builtin(__builtin_amdgcn_s_wait_tensorcnt)
#define USE_TDM 1
#else
#define USE_TDM 0
#endif

// =====================================================================
// Kernel 0: zero the stats accumulators (ws is poisoned by the harness)
// =====================================================================
__global__ void zero_stats_kernel(float* __restrict__ stats) {
    if (threadIdx.x < 256) stats[threadIdx.x] = 0.0f;
}

// =====================================================================
// Kernel 1: furthest point sampling. One 1024-thread block per batch.
// Points live in registers (8/thread); 2047 serial steps with a wave32
// shuffle + LDS argmax tree; the winning thread publishes the next
// centroid from its own registers (no global load on the serial chain).
// =====================================================================
__global__ __launch_bounds__(1024)
void fps_kernel(const float* __restrict__ pts, float* __restrict__ newp) {
    const int b   = blockIdx.x;
    const int tid = threadIdx.x;
    const float* p = pts + (size_t)b * N_PER * 3;

    float px[8], py[8], pz[8], md[8];
#pragma unroll
    for (int i = 0; i < 8; ++i) {
        int idx = tid + i * 1024;
        px[i] = p[idx * 3 + 0];
        py[i] = p[idx * 3 + 1];
        pz[i] = p[idx * 3 + 2];
        md[i] = 1e10f;
    }

    __shared__ float cur[3];
    __shared__ unsigned long long red[32];
    __shared__ unsigned long long winner;

    if (tid == 0) {
        cur[0] = px[0]; cur[1] = py[0]; cur[2] = pz[0];
        newp[((size_t)b * M_PER) * 3 + 0] = px[0];
        newp[((size_t)b * M_PER) * 3 + 1] = py[0];
        newp[((size_t)b * M_PER) * 3 + 2] = pz[0];
    }
    __syncthreads();

    for (int j = 1; j < M_PER; ++j) {
        const float cx = cur[0], cy = cur[1], cz = cur[2];
        float bestv = -1.0f;
        int   besti = 0;
#pragma unroll
        for (int i = 0; i < 8; ++i) {
            float dx = px[i] - cx, dy = py[i] - cy, dz = pz[i] - cz;
            float d  = dx * dx + dy * dy + dz * dz;
            md[i] = fminf(md[i], d);
            if (md[i] > bestv) { bestv = md[i]; besti = tid + i * 1024; }
        }
        unsigned long long pk =
            ((unsigned long long)__float_as_uint(bestv) << 32) | (unsigned)besti;
#pragma unroll
        for (int off = 16; off > 0; off >>= 1) {
            unsigned long long o = __shfl_xor(pk, off, 32);
            pk = (o > pk) ? o : pk;
        }
        if ((tid & 31) == 0) red[tid >> 5] = pk;
        __syncthreads();
        if (tid < 32) {
            unsigned long long v = red[tid];
#pragma unroll
            for (int off = 16; off > 0; off >>= 1) {
                unsigned long long o = __shfl_xor(v, off, 32);
                v = (o > v) ? o : v;
            }
            if (tid == 0) winner = v;
        }
        __syncthreads();
        unsigned widx = (unsigned)(winner & 0xffffffffu);
        if ((widx & 1023u) == (unsigned)tid) {
            int slot = (int)(widx >> 10);
            cur[0] = px[slot]; cur[1] = py[slot]; cur[2] = pz[slot];
            size_t o = ((size_t)b * M_PER + j) * 3;
            newp[o + 0] = px[slot];
            newp[o + 1] = py[slot];
            newp[o + 2] = pz[slot];
        }
        __syncthreads();
    }
}

// =====================================================================
// Kernel 2: brute-force KNN (k=16). One wave32 per query, 8 queries per
// 256-thread block. Point tiles staged into LDS by the Tensor Data
// Mover (double-buffered, s_wait_tensorcnt pipelining); per-lane top-16
// in registers, then 16 register-only selection rounds via shuffles.
// =====================================================================
#define KTILE 2048
#define KT_DW (KTILE * 3)       // dwords per tile = 6144

#if USE_TDM
// Issue one TDM load of KT_DW contiguous dwords (global -> LDS).
// D# per CDNA5 ISA ch.8: group0 {count, lds_addr, global_addr, type=2},
// group1 {data_size=4B, tensor_dim0=tile_dim0=KT_DW, (tile_)dim1=1}.
__device__ __forceinline__ void tdm_load_tile(const float* gsrc, unsigned lds_off) {
    unsigned long long ga = (unsigned long long)(size_t)gsrc;
    v4u g0;
    g0.x = 1u;                                     // count=1 (valid descriptor)
    g0.y = lds_off;                                // lds_addr (bytes)
    g0.z = (unsigned)ga;                           // global_addr[31:0]
    g0.w = (unsigned)((ga >> 32) & 0x1FFFFFFu)     // global_addr[56:32]
         | (2u << 30);                             // type = 2 ("image")
    v8i g1;
    g1[0] = 0x20000;                 // workgroup_mask=0, data_size=2 (4B)
    g1[1] = (int)(KT_DW << 16);      // tensor_dim0[15:0] in bits[63:48]
    g1[2] = 0x10000;                 // tensor_dim0[31:16]=0, tensor_dim1[15:0]=1
    g1[3] = (int)(KT_DW << 16);      // tensor_dim1[31:16]=0, tile_dim0=KT_DW
    g1[4] = 1;                       // tile_dim1=1, tile_dim2=0
    g1[5] = KT_DW;                   // tensor_dim0_stride[31:0]
    g1[6] = (int)(KT_DW << 16);      // dim0_stride[47:32]=0, dim1_stride[15:0]
    g1[7] = 0;
    v4i z4 = {0, 0, 0, 0};
    v8i z8 = {0, 0, 0, 0, 0, 0, 0, 0};
    // clang-23 / therock form: (g0, g1, g2, g3, <extra group>, cpol)
    __builtin_amdgcn_tensor_load_to_lds(g0, g1, z4, z4, z8, 0);
}
#endif

__global__ __launch_bounds__(256)
void knn_kernel(const float* __restrict__ pts,
                const float* __restrict__ newp,
                int* __restrict__ nidx) {
#if USE_TDM
    __shared__ float tp[2][KT_DW];           // 48 KB, double buffered
#else
    __shared__ float tp[1][KT_DW];           // 24 KB
#endif

    const int blocks_per_batch = M_PER / 8;  // 256
    const int b     = blockIdx.x / blocks_per_batch;
    const int qbase = (blockIdx.x % blocks_per_batch) * 8;
    const int wave  = threadIdx.x >> 5;
    const int lane  = threadIdx.x & 31;
    const int q     = qbase + wave;
    const size_t qg = (size_t)b * M_PER + q;

    const float qx = newp[qg * 3 + 0];
    const float qy = newp[qg * 3 + 1];
    const float qz = newp[qg * 3 + 2];

    float d[KNN];
    int   id[KNN];
#pragma unroll
    for (int i = 0; i < KNN; ++i) { d[i] = 1e30f; id[i] = 0; }
    float worst = 1e30f;
    int   wslot = 0;

    const float* p = pts + (size_t)b * N_PER * 3;
    const int nt = N_PER / KTILE;            // 4 tiles

#if USE_TDM
    if (wave == 0)
        tdm_load_tile(p, (unsigned)(size_t)(void*)&tp[0][0]);
#endif

    for (int t = 0; t < nt; ++t) {
#if USE_TDM
        const float* buf = tp[t & 1];
        if (t + 1 < nt) {
            if (wave == 0)
                tdm_load_tile(p + (size_t)(t + 1) * KT_DW,
                              (unsigned)(size_t)(void*)&tp[(t + 1) & 1][0]);
            __builtin_amdgcn_s_wait_tensorcnt((short)1);  // tile t done (in-order)
        } else {
            __builtin_amdgcn_s_wait_tensorcnt((short)0);
        }
        __syncthreads();                      // tile t visible to all waves
#else
        const float* buf = tp[0];
        __syncthreads();
        for (int i = threadIdx.x; i < KT_DW; i += 256)
            tp[0][i] = p[(size_t)t * KT_DW + i];
        if (t + 1 < nt)                       // gfx1250 global_prefetch_b8
            __builtin_prefetch(p + (size_t)(t + 1) * KT_DW + lane * 64, 0, 0);
        __syncthreads();
#endif

        for (int i = lane; i < KTILE; i += 32) {
            float dx = buf[i * 3 + 0] - qx;
            float dy = buf[i * 3 + 1] - qy;
            float dz = buf[i * 3 + 2] - qz;
            float dist = dx * dx + dy * dy + dz * dz;
            int   gi   = t * KTILE + i;
            if (dist < worst) {
#pragma unroll
                for (int s = 0; s < KNN; ++s)
                    if (s == wslot) { d[s] = dist; id[s] = gi; }
                worst = -1.0f;
#pragma unroll
                for (int s = 0; s < KNN; ++s)
                    if (d[s] > worst) { worst = d[s]; wslot = s; }
            }
        }
#if USE_TDM
        __syncthreads();                      // consumption done before reuse
#endif
    }

    // merge 32 lanes x 16 candidates -> 16 smallest, registers only
    for (int r = 0; r < KNN; ++r) {
        unsigned long long best = ~0ull;
        int bslot = -1;
#pragma unroll
        for (int s = 0; s < KNN; ++s) {
            unsigned long long v =
                ((unsigned long long)__float_as_uint(d[s]) << 32) | (unsigned)id[s];
            if (v < best) { best = v; bslot = s; }
        }
        unsigned long long wb = best;
#pragma unroll
        for (int off = 16; off > 0; off >>= 1) {
            unsigned long long o = __shfl_xor(wb, off, 32);
            wb = (o < wb) ? o : wb;
        }
        if (wb == best) {
#pragma unroll
            for (int s = 0; s < KNN; ++s)
                if (s == bslot) d[s] = 1e30f;
        }
        if (lane == 0) nidx[qg * KNN + r] = (int)(wb & 0x7fffffffu);
    }
}

// =====================================================================
// Kernels 3 & 5: fused gather + fp32 WMMA GEMM.
//   One 16x16 tile = 16 neighbors of one sampled point x 16 output
//   channels; 8 waves cover all 128 channels and share the gathered
//   A-tile in (double-buffered) LDS. B fragments (W) stay in registers
//   for all MPB m-iterations. PASS 0 accumulates BN sum/sumsq in
//   registers and flushes once; PASS 1 fuses normalize+ReLU+maxpool.
// =====================================================================
template <int PASS>
__global__ __launch_bounds__(256)
void gemm_kernel(const float* __restrict__ pts,
                 const float* __restrict__ feat,
                 const float* __restrict__ newp,
                 const int*   __restrict__ nidx,
                 const float* __restrict__ W,
                 float*       __restrict__ stats,
                 const float* __restrict__ scale,
                 const float* __restrict__ bias,
                 float*       __restrict__ outf) {
    __shared__ float g[2][16][C_PAD];       // 8.5 KB, double buffered

    const int wave = threadIdx.x >> 5;      // ctile 0..7
    const int lane = threadIdx.x & 31;
    const int row  = lane & 15;
    const int kb   = (lane >> 4) << 1;      // 0 or 2

    // ---- B fragments (W transposed view) into registers ----
    v2f bfrag[NCHUNK];
    {
        const int o = wave * 16 + row;      // output channel
#pragma unroll
        for (int ch = 0; ch < NCHUNK; ++ch) {
            int c0 = ch * 4 + kb;
            bfrag[ch].x = (c0     < C_G) ? W[o * C_G + c0]     : 0.0f;
            bfrag[ch].y = (c0 + 1 < C_G) ? W[o * C_G + c0 + 1] : 0.0f;
        }
    }

    const int mBase = blockIdx.x * MPB;

    // gather A-tile for sampled point m into LDS buffer bi
    auto gather = [&](int m, int bi) {
        const int bb = m / M_PER;
        const float* p = pts  + (size_t)bb * N_PER * 3;
        const float* f = feat + (size_t)bb * N_PER * C_IN;
        const float nx = newp[(size_t)m * 3 + 0];
        const float ny = newp[(size_t)m * 3 + 1];
        const float nz = newp[(size_t)m * 3 + 2];
#pragma unroll
        for (int rr = 0; rr < 2; ++rr) {
            const int k  = wave * 2 + rr;
            const int nb = nidx[(size_t)m * KNN + k];
#pragma unroll
            for (int cc = 0; cc < 3; ++cc) {
                const int c = lane + cc * 32;
                if (c < C_PAD) {
                    float v;
                    if (c < 3)
                        v = p[(size_t)nb * 3 + c] - (c == 0 ? nx : (c == 1 ? ny : nz));
                    else if (c < C_G)
                        v = f[(size_t)nb * C_IN + (c - 3)];
                    else
                        v = 0.0f;
                    g[bi][k][c] = v;
                }
            }
        }
    };

    float sAcc = 0.0f, s2Acc = 0.0f;        // PASS 0 register accumulators
    float sc = 0.0f, bi_ = 0.0f;            // PASS 1 per-channel affine
    if (PASS == 1) {
        const int ch = wave * 16 + row;
        sc  = scale[ch];
        bi_ = bias[ch];
    }

    gather(mBase, 0);

    for (int i = 0; i < MPB; ++i) {
        const int m = mBase + i;
        __syncthreads();                    // buf i&1 ready; buf (i+1)&1 consumed

        v8f acc = {0.f, 0.f, 0.f, 0.f, 0.f, 0.f, 0.f, 0.f};
#pragma unroll
        for (int ch = 0; ch < NCHUNK; ++ch) {
            v2f a;
            a.x = g[i & 1][row][ch * 4 + kb];
            a.y = g[i & 1][row][ch * 4 + kb + 1];
            acc = __builtin_amdgcn_wmma_f32_16x16x4_f32(
                false, a, false, bfrag[ch], (short)0, acc, false, false);
        }

        if (i + 1 < MPB) gather(m + 1, (i + 1) & 1);

        if (PASS == 0) {
#pragma unroll
            for (int j = 0; j < 8; ++j) { float v = acc[j]; sAcc += v; s2Acc += v * v; }
        } else {
            float mx = -1e30f;
#pragma unroll
            for (int j = 0; j < 8; ++j)
                mx = fmaxf(mx, fmaxf(acc[j] * sc + bi_, 0.0f));
            mx = fmaxf(mx, __shfl_xor(mx, 16, 32));
            if (lane < 16) outf[(size_t)m * C_OUT + wave * 16 + row] = mx;
        }
    }

    if (PASS == 0) {
        sAcc  += __shfl_xor(sAcc, 16, 32);
        s2Acc += __shfl_xor(s2Acc, 16, 32);
        if (lane < 16) {
            atomicAdd(&stats[wave * 16 + lane], sAcc);
            atomicAdd(&stats[128 + wave * 16 + lane], s2Acc);
        }
    }
}

// =====================================================================
// Kernel 4: finalize BN stats -> per-channel scale/bias; write row_splits
// =====================================================================
__global__ void finalize_kernel(const float* __restrict__ stats,
                                const float* __restrict__ gamma,
                                const float* __restrict__ beta,
                                float* __restrict__ scale,
                                float* __restrict__ bias,
                                long long* __restrict__ rs) {
    const int c = threadIdx.x;
    if (c < C_OUT) {
        const float cnt  = (float)(M_TOT * KNN);
        const float mean = stats[c] / cnt;
        const float var  = stats[128 + c] / cnt - mean * mean;
        const float s    = gamma[c] * rsqrtf(var + EPSBN);
        scale[c] = s;
        bias[c]  = beta[c] - mean * s;
    }
    if (c <= BATCH) rs[c] = (long long)c * M_PER;
}

// =====================================================================
extern "C" void kernel_launch(void* const* d_in, const int* in_sizes, int n_in,
                              void* d_out, int out_size, void* d_ws, size_t ws_size,
                              hipStream_t stream) {
    (void)in_sizes; (void)n_in; (void)out_size; (void)ws_size;
    const float* point = (const float*)d_in[0];
    const float* feat  = (const float*)d_in[1];
    // d_in[2] = row_splits (int64) -- statically known, unused
    const float* W     = (const float*)d_in[3];
    const float* gamma = (const float*)d_in[4];
    const float* beta  = (const float*)d_in[5];

    float* out   = (float*)d_out;
    float* newp  = out + OUT_NEWP_OFF;
    float* outf  = out + OUT_FEAT_OFF;
    long long* rs = (long long*)(out + OUT_RS_OFF);

    char*  ws    = (char*)d_ws;
    int*   nidx  = (int*)(ws + WS_NIDX_OFF);
    float* stats = (float*)(ws + WS_STATS_OFF);
    float* scale = (float*)(ws + WS_SCALE_OFF);
    float* bias  = (float*)(ws + WS_BIAS_OFF);

    zero_stats_kernel<<<1, 256, 0, stream>>>(stats);
    fps_kernel<<<BATCH, 1024, 0, stream>>>(point, newp);
    knn_kernel<<<M_TOT / 8, 256, 0, stream>>>(point, newp, nidx);
    gemm_kernel<0><<<GEMM_GRID, 256, 0, stream>>>(point, feat, newp, nidx, W,
                                                  stats, scale, bias, outf);
    finalize_kernel<<<1, 256, 0, stream>>>(stats, gamma, beta, scale, bias, rs);
    gemm_kernel<1><<<GEMM_GRID, 256, 0, stream>>>(point, feat, newp, nidx, W,
                                                  stats, scale, bias, outf);
}